// SwapCountLoss_4879082849029
// MI455X (gfx1250) — compile-verified
//
#include <hip/hip_runtime.h>
#include <hip/hip_bf16.h>

// Problem constants (match reference)
#define B_DIM 32
#define Q_DIM 256      // Q_LOG == Q_PHYS == 256
#define E_DIM 4096
#define CHUNK 32       // q-columns per block
#define NCHUNK (Q_DIM / CHUNK)          // 8
#define LDS_STRIDE 33                   // padded to kill bank conflicts
#define NTHREADS 512                    // 16 waves (wave32)
#define NWAVES (NTHREADS / 32)

typedef __attribute__((ext_vector_type(2))) float v2f;
typedef __attribute__((ext_vector_type(8))) float v8f;

__device__ __forceinline__ float dswap(float x) {
    // 3 * max(d_hw - 1, 0)
    return 3.0f * fmaxf(x - 1.0f, 0.0f);
}

__global__ void swap_loss_init(float* ws_num) {
    ws_num[0] = 0.0f;
}

// One block per (q-chunk, batch). Computes Mc = P[b] @ D[:, qbase:qbase+32]
// (256x32, K=256) via V_WMMA_F32_16X16X4_F32 into LDS, then accumulates
// sum_e w_e * sum_{q in chunk} Mc[i_e, q] * P[b, j_e, qbase+q] into ws_num.
__global__ void __launch_bounds__(NTHREADS)
swap_loss_main(const float* __restrict__ P,
               const float* __restrict__ d_hw,
               const int* __restrict__ edge_i,
               const int* __restrict__ edge_j,
               const float* __restrict__ edge_w,
               float* __restrict__ ws_num) {
    __shared__ float Mc[Q_DIM * LDS_STRIDE];   // 33792 bytes

    const int chunk = blockIdx.x;              // 0..7
    const int b     = blockIdx.y;              // 0..31
    const int qbase = chunk * CHUNK;

    const int tid    = threadIdx.x;
    const int wave   = tid >> 5;               // 0..15 -> row tile
    const int lane   = tid & 31;
    const int lane15 = lane & 15;
    const int half   = lane >> 4;              // 0: K+0/1, 1: K+2/3

    // ---- GEMM phase: wave w computes rows [w*16, w*16+16) x 32 cols ----
    const int r0 = wave * 16;
    // A operand: lane (low/high half) holds P[b][r0+lane15][k + 2*half + {0,1}]
    const float* Arow = P + ((size_t)b * Q_DIM + (r0 + lane15)) * Q_DIM + 2 * half;
    // B operand: lane holds D[k + 2*half + {0,1}][qbase + c0 + lane15]
    const float* Bcol0 = d_hw + (size_t)(2 * half) * Q_DIM + (qbase + lane15);
    const float* Bcol1 = Bcol0 + 16;

    v8f c0 = {}, c1 = {};
#pragma unroll 8
    for (int k = 0; k < Q_DIM; k += 4) {
        v2f a;
        a.x = Arow[k + 0];
        a.y = Arow[k + 1];
        v2f b0, b1;
        const float* bp = Bcol0 + (size_t)k * Q_DIM;
        b0.x = dswap(bp[0]);
        b0.y = dswap(bp[Q_DIM]);
        bp = Bcol1 + (size_t)k * Q_DIM;
        b1.x = dswap(bp[0]);
        b1.y = dswap(bp[Q_DIM]);
        // D = A*B + C, full f32 matrix pipe
        c0 = __builtin_amdgcn_wmma_f32_16x16x4_f32(false, a, false, b0,
                                                   (short)0, c0, false, false);
        c1 = __builtin_amdgcn_wmma_f32_16x16x4_f32(false, a, false, b1,
                                                   (short)0, c1, false, false);
    }

    // C/D layout: lanes 0-15: VGPR r -> M=r, N=lane; lanes 16-31: M=r+8, N=lane-16
    union { v8f v; float f[8]; } u0, u1;
    u0.v = c0; u1.v = c1;
#pragma unroll
    for (int r = 0; r < 8; ++r) {
        int row = r0 + r + 8 * half;
        Mc[row * LDS_STRIDE + lane15]      = u0.f[r];
        Mc[row * LDS_STRIDE + 16 + lane15] = u1.f[r];
    }
    __syncthreads();

    // ---- Edge phase: thread owns 8 edges, partial dot over this q-chunk ----
    const float* Pb = P + (size_t)b * Q_DIM * Q_DIM + qbase;
    float total = 0.0f;
#pragma unroll
    for (int n = 0; n < E_DIM / NTHREADS; ++n) {
        int e = tid + n * NTHREADS;
        int i = edge_i[e];
        int j = edge_j[e];
        float w = edge_w[e];
        const float* mrow = Mc + i * LDS_STRIDE;
        const float* prow = Pb + (size_t)j * Q_DIM;
        float s = 0.0f;
#pragma unroll
        for (int q = 0; q < CHUNK; q += 4) {
            float4 p = *(const float4*)(prow + q);
            s += mrow[q + 0] * p.x;
            s += mrow[q + 1] * p.y;
            s += mrow[q + 2] * p.z;
            s += mrow[q + 3] * p.w;
        }
        total += w * s;
    }

    // wave32 shuffle reduce, one atomic per wave
#pragma unroll
    for (int off = 16; off > 0; off >>= 1)
        total += __shfl_down(total, off, 32);
    if (lane == 0)
        atomicAdd(ws_num, total);
}

__global__ void swap_loss_finalize(const float* __restrict__ edge_w,
                                   const float* __restrict__ ws_num,
                                   float* __restrict__ out) {
    __shared__ float red[256];
    int tid = threadIdx.x;
    float s = 0.0f;
    for (int e = tid; e < E_DIM; e += 256) s += edge_w[e];
    red[tid] = s;
    __syncthreads();
    for (int st = 128; st > 0; st >>= 1) {
        if (tid < st) red[tid] += red[tid + st];
        __syncthreads();
    }
    if (tid == 0) {
        float tw = fmaxf(red[0], 1e-8f);
        out[0] = ws_num[0] / ((float)B_DIM * tw);
    }
}

extern "C" void kernel_launch(void* const* d_in, const int* in_sizes, int n_in,
                              void* d_out, int out_size, void* d_ws, size_t ws_size,
                              hipStream_t stream) {
    const float* P      = (const float*)d_in[0];
    const float* d_hw   = (const float*)d_in[1];
    const int*   edge_i = (const int*)d_in[2];
    const int*   edge_j = (const int*)d_in[3];
    const float* edge_w = (const float*)d_in[4];
    float* out    = (float*)d_out;
    float* ws_num = (float*)d_ws;   // one float of scratch

    swap_loss_init<<<1, 1, 0, stream>>>(ws_num);
    dim3 grid(NCHUNK, B_DIM);       // 8 x 32 = 256 blocks
    swap_loss_main<<<grid, NTHREADS, 0, stream>>>(P, d_hw, edge_i, edge_j,
                                                  edge_w, ws_num);
    swap_loss_finalize<<<1, 256, 0, stream>>>(edge_w, ws_num, out);
}